// CriticNetwork_47012712022392
// MI455X (gfx1250) — compile-verified
//
#include <hip/hip_runtime.h>
#include <math.h>

typedef __attribute__((ext_vector_type(2))) float v2f;
typedef __attribute__((ext_vector_type(8))) float v8f;

#define NN 50000
#define NE 800000
#define DIN 128
#define HD 64
#define POOL_BLOCKS 240

// ---------------- degree / norm ----------------
__global__ void init_deg_kernel(float* __restrict__ deg, int n) {
    int i = blockIdx.x * blockDim.x + threadIdx.x;
    if (i < n) deg[i] = 1.0f;  // self-loop
}

__global__ void deg_count_kernel(const int* __restrict__ dst, float* __restrict__ deg, int e) {
    int i = blockIdx.x * blockDim.x + threadIdx.x;
    if (i < e) atomicAdd(&deg[dst[i]], 1.0f);
}

__global__ void rsqrt_kernel(float* __restrict__ deg, int n) {
    int i = blockIdx.x * blockDim.x + threadIdx.x;
    if (i < n) deg[i] = rsqrtf(deg[i]);
}

// ---------------- fp32 WMMA GEMM: out[n,64] = act(A[n,K]) @ W[K,64] ----------------
// One wave32 computes one 16x16 output tile via V_WMMA_F32_16X16X4_F32.
// n must be a multiple of 16 (50000 = 3125*16). K and RELU are compile-time so the
// inner loop is a clean load -> (optional single v_max) -> v_wmma pipeline.
template <int K, bool RELU>
__global__ __launch_bounds__(256) void gemm_wmma_f32(
    const float* __restrict__ A, const float* __restrict__ W,
    float* __restrict__ out, int n) {
    const int lane = threadIdx.x;        // 0..31
    const int wave = threadIdx.y;        // 0..7
    const int ntn = HD / 16;             // 4 column tiles
    const int tiles_total = (n / 16) * ntn;
    int tile = blockIdx.x * 8 + wave;    // wave-uniform
    if (tile >= tiles_total) return;     // wave-uniform exit: EXEC stays all-1s

    const int tm = tile / ntn;
    const int tn = tile % ntn;
    const int half = lane >> 4;          // 0: lanes 0-15, 1: lanes 16-31
    const int l15  = lane & 15;
    const int row  = tm * 16 + l15;      // A-frag: M = lane%16
    const int col  = tn * 16 + l15;      // B-frag: N = lane%16
    const int khalf = half * 2;          // K sub-offset per half-wave

    const float* arow = A + (size_t)row * K;
    v8f acc = {};
#pragma unroll 8
    for (int k0 = 0; k0 < K; k0 += 4) {
        const int kb = k0 + khalf;
        v2f a, b;
        a.x = arow[kb];
        a.y = arow[kb + 1];
        if (RELU) { a.x = fmaxf(a.x, 0.0f); a.y = fmaxf(a.y, 0.0f); }
        b.x = W[(size_t)kb * HD + col];
        b.y = W[(size_t)(kb + 1) * HD + col];
        // 8 args: (neg_a, A, neg_b, B, c_mod, C, reuse_a, reuse_b)
        acc = __builtin_amdgcn_wmma_f32_16x16x4_f32(
            false, a, false, b, (short)0, acc, false, false);
    }
    // C/D layout: lanes 0-15 -> M = tm*16 + r, lanes 16-31 -> M = tm*16 + 8 + r; N = col
    const int mbase = tm * 16 + half * 8;
#pragma unroll
    for (int r = 0; r < 8; ++r) {
        out[(size_t)(mbase + r) * HD + col] = acc[r];
    }
}

// ---------------- agg = dinv[i]^2 * h[i] + b  (pre-relu; edge atomics add on top) ----
__global__ void selfloop_bias_kernel(const float* __restrict__ h,
                                     const float* __restrict__ dinv,
                                     const float* __restrict__ b,
                                     float* __restrict__ agg, int n) {
    int t = blockIdx.x * blockDim.x + threadIdx.x;   // n*16 threads, float4 each
    int node = t >> 4;
    int c4 = (t & 15) * 4;
    if (node >= n) return;
    float s = dinv[node];
    s = s * s;
    const float4 hv = *(const float4*)(h + (size_t)node * HD + c4);
    float4 o;
    o.x = s * hv.x + b[c4 + 0];
    o.y = s * hv.y + b[c4 + 1];
    o.z = s * hv.z + b[c4 + 2];
    o.w = s * hv.w + b[c4 + 3];
    *(float4*)(agg + (size_t)node * HD + c4) = o;
}

// ---------------- edge scatter: agg[dst] += norm * h[src] ----------------
// 16 lanes per edge, float4 per lane: half-wave reads one 256B node row coalesced.
__global__ void edge_scatter_kernel(const float* __restrict__ h,
                                    const int* __restrict__ src,
                                    const int* __restrict__ dst,
                                    const float* __restrict__ dinv,
                                    float* __restrict__ agg, int e) {
    int t = blockIdx.x * blockDim.x + threadIdx.x;
    int ed = t >> 4;
    if (ed >= e) return;
    int c4 = (t & 15) * 4;
    int s = src[ed];
    int d = dst[ed];
    float norm = dinv[s] * dinv[d];
    const float4 hv = *(const float4*)(h + (size_t)s * HD + c4);
    float* ab = agg + (size_t)d * HD + c4;
    atomicAdd(ab + 0, hv.x * norm);
    atomicAdd(ab + 1, hv.y * norm);
    atomicAdd(ab + 2, hv.z * norm);
    atomicAdd(ab + 3, hv.w * norm);
}

// ---------------- pooling: per-block partial sum & max of relu(h) per column ---------
__global__ __launch_bounds__(256) void pool_partial_kernel(const float* __restrict__ h,
                                                           float* __restrict__ psum,
                                                           float* __restrict__ pmax, int n) {
    const int c = threadIdx.x & 63;      // column
    const int rg = threadIdx.x >> 6;     // row group 0..3
    float s = 0.0f, m = 0.0f;            // relu output >= 0
    for (int row = blockIdx.x * 4 + rg; row < n; row += gridDim.x * 4) {
        float v = fmaxf(h[(size_t)row * HD + c], 0.0f);  // relu of layer-3 output
        s += v;
        m = fmaxf(m, v);
    }
    __shared__ float ls[256];
    __shared__ float lm[256];
    ls[threadIdx.x] = s;
    lm[threadIdx.x] = m;
    __syncthreads();
    if (rg == 0) {
        s = ls[c] + ls[64 + c] + ls[128 + c] + ls[192 + c];
        m = fmaxf(fmaxf(lm[c], lm[64 + c]), fmaxf(lm[128 + c], lm[192 + c]));
        psum[blockIdx.x * 64 + c] = s;
        pmax[blockIdx.x * 64 + c] = m;
    }
}

// ---------------- final reduce + MLP head: [1,128] @ [128,64] relu @ [64,1] ----------
__global__ __launch_bounds__(64) void head_kernel(const float* __restrict__ psum,
                                                  const float* __restrict__ pmax,
                                                  int nblocks,
                                                  const float* __restrict__ fw1,
                                                  const float* __restrict__ fb1,
                                                  const float* __restrict__ fw2,
                                                  const float* __restrict__ fb2,
                                                  float* __restrict__ out, int n) {
    __shared__ float pooled[2 * HD];
    __shared__ float red[HD];
    const int j = threadIdx.x;  // 0..63
    float s = 0.0f, m = 0.0f;
    for (int b = 0; b < nblocks; ++b) {
        s += psum[b * 64 + j];
        m = fmaxf(m, pmax[b * 64 + j]);
    }
    pooled[j] = s / (float)n;   // mean
    pooled[HD + j] = m;         // max
    __syncthreads();
    float hv = fb1[j];
    for (int i = 0; i < 2 * HD; ++i) hv += pooled[i] * fw1[(size_t)i * HD + j];
    hv = fmaxf(hv, 0.0f);
    red[j] = hv * fw2[j];
    __syncthreads();
    for (int st = 32; st > 0; st >>= 1) {
        if (j < st) red[j] += red[j + st];
        __syncthreads();
    }
    if (j == 0) out[0] = red[0] + fb2[0];
}

extern "C" void kernel_launch(void* const* d_in, const int* in_sizes, int n_in,
                              void* d_out, int out_size, void* d_ws, size_t ws_size,
                              hipStream_t stream) {
    const float* x   = (const float*)d_in[0];
    const int*   ei  = (const int*)d_in[1];     // [2, E]: row0 = src, row1 = dst
    const float* W1  = (const float*)d_in[2];
    const float* b1  = (const float*)d_in[3];
    const float* W2  = (const float*)d_in[4];
    const float* b2  = (const float*)d_in[5];
    const float* W3  = (const float*)d_in[6];
    const float* b3  = (const float*)d_in[7];
    const float* fw1 = (const float*)d_in[8];
    const float* fb1 = (const float*)d_in[9];
    const float* fw2 = (const float*)d_in[10];
    const float* fb2 = (const float*)d_in[11];
    float* out = (float*)d_out;

    const int* src = ei;
    const int* dst = ei + NE;

    // Workspace layout (~26 MB)
    float* dinv = (float*)d_ws;               // NN
    float* hA   = dinv + NN;                  // NN*HD
    float* hB   = hA + (size_t)NN * HD;       // NN*HD
    float* psum = hB + (size_t)NN * HD;       // POOL_BLOCKS*64
    float* pmax = psum + POOL_BLOCKS * 64;    // POOL_BLOCKS*64

    const int T = 256;
    const int gemm_tiles = (NN / 16) * (HD / 16);         // 12500
    const int gemm_blocks = (gemm_tiles + 7) / 8;
    dim3 gemm_block(32, 8);
    const int node_blk = (NN + T - 1) / T;
    const int edge_blk = (NE + T - 1) / T;
    const int node16_blk = (NN * 16 + T - 1) / T;
    const int edge16_blk = (NE * 16 + T - 1) / T;

    // degrees -> dinv
    init_deg_kernel<<<node_blk, T, 0, stream>>>(dinv, NN);
    deg_count_kernel<<<edge_blk, T, 0, stream>>>(dst, dinv, NE);
    rsqrt_kernel<<<node_blk, T, 0, stream>>>(dinv, NN);

    // Layer 1: hA = x@W1 ; hB = dinv^2*hA + b1 + scatter
    gemm_wmma_f32<DIN, false><<<gemm_blocks, gemm_block, 0, stream>>>(x, W1, hA, NN);
    selfloop_bias_kernel<<<node16_blk, T, 0, stream>>>(hA, dinv, b1, hB, NN);
    edge_scatter_kernel<<<edge16_blk, T, 0, stream>>>(hA, src, dst, dinv, hB, NE);

    // Layer 2: hA = relu(hB)@W2 ; hB = dinv^2*hA + b2 + scatter
    gemm_wmma_f32<HD, true><<<gemm_blocks, gemm_block, 0, stream>>>(hB, W2, hA, NN);
    selfloop_bias_kernel<<<node16_blk, T, 0, stream>>>(hA, dinv, b2, hB, NN);
    edge_scatter_kernel<<<edge16_blk, T, 0, stream>>>(hA, src, dst, dinv, hB, NE);

    // Layer 3
    gemm_wmma_f32<HD, true><<<gemm_blocks, gemm_block, 0, stream>>>(hB, W3, hA, NN);
    selfloop_bias_kernel<<<node16_blk, T, 0, stream>>>(hA, dinv, b3, hB, NN);
    edge_scatter_kernel<<<edge16_blk, T, 0, stream>>>(hA, src, dst, dinv, hB, NE);

    // Pool (relu applied on read) + head
    pool_partial_kernel<<<POOL_BLOCKS, T, 0, stream>>>(hB, psum, pmax, NN);
    head_kernel<<<1, 64, 0, stream>>>(psum, pmax, POOL_BLOCKS, fw1, fb1, fw2, fb2, out, NN);
}